// GraphAttentionLayer_6949257085380
// MI455X (gfx1250) — compile-verified
//
#include <hip/hip_runtime.h>
#include <hip/hip_bf16.h>
#include <stdint.h>

#define B_  8
#define V_  8
#define F_  32
#define T_  4
#define P_  4
#define K_  3
#define R_  12
#define D_  32
#define NO_ 32

// output layout (flat, return order): h_f, h_c, h_t, path_attn, attn_causal
#define HF_OFF 0
#define HC_OFF (B_*V_*D_)                 // 2048
#define HT_OFF (2*B_*V_*D_)               // 4096
#define PA_OFF (3*B_*V_*D_)               // 6144
#define AC_OFF (3*B_*V_*D_ + B_*V_*F_*T_*P_)  // 38912

typedef float v2f __attribute__((ext_vector_type(2)));
typedef float v8f __attribute__((ext_vector_type(8)));

// ---------------- JAX Threefry-2x32 (exact) ----------------
__device__ __forceinline__ unsigned rotl32(unsigned x, int r) {
    return (x << r) | (x >> (32 - r));
}
__device__ __forceinline__ void tf_round(unsigned& x0, unsigned& x1, int r) {
    x0 += x1; x1 = rotl32(x1, r); x1 ^= x0;
}
__device__ __forceinline__ void threefry2x32(unsigned k0, unsigned k1,
                                             unsigned c0, unsigned c1,
                                             unsigned& o0, unsigned& o1) {
    unsigned ks0 = k0, ks1 = k1, ks2 = k0 ^ k1 ^ 0x1BD11BDAu;
    unsigned x0 = c0 + ks0, x1 = c1 + ks1;
    tf_round(x0,x1,13); tf_round(x0,x1,15); tf_round(x0,x1,26); tf_round(x0,x1,6);
    x0 += ks1; x1 += ks2 + 1u;
    tf_round(x0,x1,17); tf_round(x0,x1,29); tf_round(x0,x1,16); tf_round(x0,x1,24);
    x0 += ks2; x1 += ks0 + 2u;
    tf_round(x0,x1,13); tf_round(x0,x1,15); tf_round(x0,x1,26); tf_round(x0,x1,6);
    x0 += ks0; x1 += ks1 + 3u;
    tf_round(x0,x1,17); tf_round(x0,x1,29); tf_round(x0,x1,16); tf_round(x0,x1,24);
    x0 += ks1; x1 += ks2 + 4u;
    tf_round(x0,x1,13); tf_round(x0,x1,15); tf_round(x0,x1,26); tf_round(x0,x1,6);
    x0 += ks2; x1 += ks0 + 5u;
    o0 = x0; o1 = x1;
}
__device__ __forceinline__ float bits_to_unit_float(unsigned bits) {
    return __uint_as_float((bits >> 9) | 0x3f800000u) - 1.0f;
}

// ---- CDNA5 async global->LDS copy (GVS mode), tracked by ASYNCcnt ----
__device__ __forceinline__ void async_g2l_b32(unsigned lds_byte_addr,
                                              unsigned voff_bytes,
                                              uint64_t sgpr_base) {
    asm volatile("global_load_async_to_lds_b32 %0, %1, %2"
                 :: "v"(lds_byte_addr), "v"(voff_bytes), "s"(sgpr_base)
                 : "memory");
}
__device__ __forceinline__ void wait_asynccnt0() {
    asm volatile("s_wait_asynccnt 0x0" ::: "memory");
}
__device__ __forceinline__ unsigned lds_addr_of(const void* p) {
    // generic pointer to __shared__: addr[31:0] is the LDS byte offset
    return (unsigned)(uintptr_t)p;
}

// ================= Kernel 1: per-edge path messages =================
// one wave32 per (b,v,f); lane == d. grid: B*V*F/8 blocks of 256 threads.
__global__ __launch_bounds__(256) void gal_edges_kernel(
    const float* __restrict__ feat,      // [B,V,F,D]
    const float* __restrict__ h_time,    // [B,V,NO]
    const float* __restrict__ rel_index, // [B,V,F,T,P,K,R]
    const float* __restrict__ params,    // [R,D,D]
    const float* __restrict__ W_p,       // [NO+D]
    const float* __restrict__ b_p,       // [1]
    float* __restrict__ out,
    float* __restrict__ m_red)           // ws: [B,V,F,D]
{
    const int lane = threadIdx.x & 31;
    // uniform wave id -> scalar (SMEM) loads for rel_index
    const int wid  = __builtin_amdgcn_readfirstlane(
                        (int)((blockIdx.x * blockDim.x + threadIdx.x) >> 5));
    const int f  = wid % F_;
    const int bv = wid / F_;
    const int d  = lane;

    // diagonal of the R relation matrices: pd[r] = params[r,d,d]
    float pd[R_];
    #pragma unroll
    for (int r = 0; r < R_; ++r) pd[r] = params[r * D_ * D_ + d * D_ + d];

    const float fe   = feat[(bv * F_ + f) * D_ + d];
    const float wsum = W_p[NO_ + d];

    // hsum = sum_a h_time[b,v,a] * W_p[a]   (wave reduction, broadcast)
    float hpart = h_time[bv * NO_ + d] * W_p[d];
    #pragma unroll
    for (int o = 16; o > 0; o >>= 1) hpart += __shfl_xor(hpart, o, 32);
    const float bias = b_p[0];

    // Gumbel noise: element e = wid*16 + lane (lanes 0..15 hold the 16 (t,p))
    // JAX uniform(key(42), (32768,)): counters split [0:16384],[16384:32768]
    unsigned e = (unsigned)(wid * 16 + lane);
    unsigned c0, c1; int pick;
    if (e < 16384u) { c0 = e;           c1 = e + 16384u; pick = 0; }
    else            { c0 = e - 16384u;  c1 = e;          pick = 1; }
    unsigned o0, o1; threefry2x32(0u, 42u, c0, c1, o0, o1);
    float u = bits_to_unit_float(pick ? o1 : o0);
    float gum = -logf(-logf(u + 1e-10f) + 1e-10f);

    float mr = 0.0f;
    const long ebase = (long)wid * (T_ * P_ * K_ * R_);

    #pragma unroll
    for (int t = 0; t < T_; ++t) {
        float mj[P_], msg[P_];
        #pragma unroll
        for (int p = 0; p < P_; ++p) {
            const float* ri = rel_index + ebase + (long)((t * P_ + p) * K_) * R_;
            float prod = 1.0f;
            #pragma unroll
            for (int k = 0; k < K_; ++k) {
                float s = 0.0f;
                #pragma unroll
                for (int r = 0; r < R_; ++r) s = fmaf(ri[k * R_ + r], pd[r], s);
                prod *= s;
            }
            mj[p] = fe * prod;
            float m = mj[p] * wsum;
            #pragma unroll
            for (int o = 16; o > 0; o >>= 1) m += __shfl_xor(m, o, 32);
            msg[p] = hpart + m + bias;   // uniform across wave
        }
        // hard gumbel argmax over p of sigmoid(msg)+g
        int pstar = 0; float best = -3.0e38f;
        #pragma unroll
        for (int p = 0; p < P_; ++p) {
            float gp = __shfl(gum, t * 4 + p, 32);
            float score = 1.0f / (1.0f + expf(-msg[p])) + gp;
            if (score > best) { best = score; pstar = p; }
        }
        // path_attn = softmax(hard_mask * msg)
        float mm[P_]; float mx = 0.0f;
        #pragma unroll
        for (int p = 0; p < P_; ++p) { mm[p] = (p == pstar) ? msg[p] : 0.0f; mx = fmaxf(mx, mm[p]); }
        float ex[P_]; float es = 0.0f;
        #pragma unroll
        for (int p = 0; p < P_; ++p) { ex[p] = expf(mm[p] - mx); es += ex[p]; }
        const float inv = 1.0f / es;
        #pragma unroll
        for (int p = 0; p < P_; ++p) {
            float pa = ex[p] * inv;          // uniform across wave
            mr = fmaf(pa, mj[p], mr);
            if (lane == p) out[PA_OFF + wid * (T_ * P_) + t * P_ + p] = pa;
        }
    }
    m_red[(bv * F_ + f) * D_ + d] = mr;
}

// ================= Kernel 2: per-(b,v) WMMA GEMMs + attention =================
// 128 threads (4 waves); each wave computes one 16x16 f32 tile via
// V_WMMA_F32_16X16X4_F32. LDS rows padded (+1 float) -> conflict-free
// A-fragment loads. Pure copies staged via GLOBAL_LOAD_ASYNC_TO_LDS_B32.
__global__ __launch_bounds__(128) void gal_bv_kernel(
    const float* __restrict__ feat,    // [B,V,F,D]
    const float* __restrict__ h_time,  // [B,V,NO]
    const float* __restrict__ cossim,  // [B,V,F,F]
    const float* __restrict__ m_red,   // ws [B,V,F,D]
    const float* __restrict__ w1,      // [D+NO, F] = [64,32]
    const float* __restrict__ b1,      // [F]
    const float* __restrict__ w2,      // [F,1]
    const float* __restrict__ b2,      // [1]
    float* __restrict__ out)
{
    __shared__ float sCos[32][33];
    __shared__ float sMred[32][33];
    __shared__ float sX[32][65];
    __shared__ float sW1[64][33];
    __shared__ float sFinal[32][33];
    __shared__ float sH1[32][33];
    __shared__ float sAttn[32];

    const int bv = blockIdx.x;
    const int v  = bv % V_;
    const int b  = bv / V_;
    const int tid = threadIdx.x;

    // ---- async stage pure copies: cos, M_red, W1  (ASYNCcnt) ----
    {
        const uint64_t cosBase  = (uint64_t)(uintptr_t)(cossim + bv * 1024);
        const uint64_t mredBase = (uint64_t)(uintptr_t)(m_red  + bv * 1024);
        const uint64_t w1Base   = (uint64_t)(uintptr_t)w1;
        for (int i = tid; i < 1024; i += 128) {
            const int r = i >> 5, c = i & 31;
            async_g2l_b32(lds_addr_of(&sCos[r][c]),  (unsigned)(i * 4), cosBase);
            async_g2l_b32(lds_addr_of(&sMred[r][c]), (unsigned)(i * 4), mredBase);
        }
        for (int i = tid; i < 2048; i += 128) {
            async_g2l_b32(lds_addr_of(&sW1[i >> 5][i & 31]), (unsigned)(i * 4), w1Base);
        }
    }
    // X = [h_time | feature_embed] needs per-element select: regular stores
    for (int i = tid; i < 2048; i += 128) {
        int f = i >> 6, a = i & 63;
        sX[f][a] = (a < 32) ? h_time[bv * NO_ + a]
                            : feat[(bv * F_ + f) * D_ + (a - 32)];
    }
    wait_asynccnt0();
    __syncthreads();

    const int w    = tid >> 5;      // wave index 0..3
    const int lane = tid & 31;
    const int m    = lane & 15;     // A row / B,C column within tile
    const int h    = lane >> 4;     // lane half
    const int fi   = (w >> 1) * 16; // C tile row base
    const int ci   = (w & 1) * 16;  // C tile col base

    // ---- h1 = relu(X[32x64] @ W1[64x32] + b1) ----
    v8f acc;
    {
        float bb = b1[ci + m];
        #pragma unroll
        for (int i = 0; i < 8; ++i) acc[i] = bb;
    }
    #pragma unroll
    for (int kk = 0; kk < 64; kk += 4) {
        v2f A, Bf;
        A.x  = sX[fi + m][kk + 2 * h + 0];
        A.y  = sX[fi + m][kk + 2 * h + 1];
        Bf.x = sW1[kk + 2 * h + 0][ci + m];
        Bf.y = sW1[kk + 2 * h + 1][ci + m];
        acc = __builtin_amdgcn_wmma_f32_16x16x4_f32(
                  false, A, false, Bf, (short)0, acc, false, false);
    }
    #pragma unroll
    for (int i = 0; i < 8; ++i) sH1[fi + i + 8 * h][ci + m] = fmaxf(acc[i], 0.0f);

    // ---- final = cos[32x32] @ M_red[32x32] ----
    v8f acc2;
    #pragma unroll
    for (int i = 0; i < 8; ++i) acc2[i] = 0.0f;
    #pragma unroll
    for (int kk = 0; kk < 32; kk += 4) {
        v2f A, Bf;
        A.x  = sCos[fi + m][kk + 2 * h + 0];
        A.y  = sCos[fi + m][kk + 2 * h + 1];
        Bf.x = sMred[kk + 2 * h + 0][ci + m];
        Bf.y = sMred[kk + 2 * h + 1][ci + m];
        acc2 = __builtin_amdgcn_wmma_f32_16x16x4_f32(
                   false, A, false, Bf, (short)0, acc2, false, false);
    }
    #pragma unroll
    for (int i = 0; i < 8; ++i) sFinal[fi + i + 8 * h][ci + m] = acc2[i];
    __syncthreads();

    // ---- attn_causal = softmax(h1 @ w2 + b2) over F (one wave) ----
    if (tid < 32) {
        const int f = tid;
        float s = b2[0];
        #pragma unroll
        for (int j = 0; j < 32; ++j) s = fmaf(sH1[f][j], w2[j], s);
        float mx = s;
        #pragma unroll
        for (int o = 16; o > 0; o >>= 1) mx = fmaxf(mx, __shfl_xor(mx, o, 32));
        float e = expf(s - mx);
        float se = e;
        #pragma unroll
        for (int o = 16; o > 0; o >>= 1) se += __shfl_xor(se, o, 32);
        float attn = e / se;
        sAttn[f] = attn;
        out[AC_OFF + bv * F_ + f] = attn;
    }
    __syncthreads();

    // ---- h_c / h_t / h_f ----
    if (tid < 32) {
        const int d = tid;
        float hc = 0.0f, tot = 0.0f;
        #pragma unroll
        for (int f2 = 0; f2 < 32; ++f2) {
            float fv = sFinal[f2][d];
            hc  = fmaf(sAttn[f2], fv, hc);
            tot += fv;
        }
        float htv = tot - hc;   // (1-attn) sum
        // II = jax.random.randint(key(7), (B,), 0, V)  ->  threefry bits & 7
        unsigned c0 = (b < 4) ? (unsigned)b       : (unsigned)(b - 4);
        unsigned c1 = (b < 4) ? (unsigned)(b + 4) : (unsigned)b;
        unsigned o0, o1; threefry2x32(0u, 7u, c0, c1, o0, o1);
        int II = (int)(((b < 4) ? o0 : o1) & 7u);
        out[HF_OFF + bv * D_ + d] = hc + ((v == II) ? htv : 0.0f);
        out[HC_OFF + bv * D_ + d] = hc;
        out[HT_OFF + bv * D_ + d] = htv;
    }
}

extern "C" void kernel_launch(void* const* d_in, const int* in_sizes, int n_in,
                              void* d_out, int out_size, void* d_ws, size_t ws_size,
                              hipStream_t stream) {
    const float* feat    = (const float*)d_in[0];
    const float* h_time  = (const float*)d_in[1];
    const float* rel_idx = (const float*)d_in[2];
    const float* cossim  = (const float*)d_in[3];
    const float* params  = (const float*)d_in[4];
    const float* W_p     = (const float*)d_in[5];
    const float* b_p     = (const float*)d_in[6];
    const float* w1      = (const float*)d_in[7];
    const float* b1      = (const float*)d_in[8];
    const float* w2      = (const float*)d_in[9];
    const float* b2      = (const float*)d_in[10];
    float* out   = (float*)d_out;
    float* m_red = (float*)d_ws;   // B*V*F*D floats = 256 KB

    // 2048 waves, 8 waves/block
    gal_edges_kernel<<<B_ * V_ * F_ / 8, 256, 0, stream>>>(
        feat, h_time, rel_idx, params, W_p, b_p, out, m_red);
    gal_bv_kernel<<<B_ * V_, 128, 0, stream>>>(
        feat, h_time, cossim, m_red, w1, b1, w2, b2, out);
}